// OTDisparityInit_30837865185322
// MI455X (gfx1250) — compile-verified
//
#include <hip/hip_runtime.h>
#include <stdint.h>

// Problem constants (from reference): B=2, D=64, H=64, W=256, C=D+W=320.
#define OT_B 2
#define OT_D 64
#define OT_H 64
#define OT_W 256
#define OT_C 320
#define OT_WP 257   // W+1 (rows incl. dustbin)
#define OT_CP 321   // C+1 (cols incl. dustbin)
#define OT_ITERS 10

typedef __attribute__((ext_vector_type(2))) float v2f;
typedef __attribute__((ext_vector_type(8))) float v8f;

// Workgroup logsumexp over up to 2 elements per thread (256 threads, 8 waves).
// Contains barriers: must be called by all threads.
__device__ __forceinline__ float wgLse2(float x0, float x1, bool has2,
                                        float* red, int wid, int lane) {
    float m = has2 ? fmaxf(x0, x1) : x0;
#pragma unroll
    for (int o = 16; o > 0; o >>= 1) m = fmaxf(m, __shfl_xor(m, o, 32));
    if (lane == 0) red[wid] = m;
    __syncthreads();
    float gm = red[0];
#pragma unroll
    for (int k = 1; k < 8; ++k) gm = fmaxf(gm, red[k]);
    float s = __expf(x0 - gm) + (has2 ? __expf(x1 - gm) : 0.0f);
#pragma unroll
    for (int o = 16; o > 0; o >>= 1) s += __shfl_xor(s, o, 32);
    if (lane == 0) red[8 + wid] = s;
    __syncthreads();
    float gs = red[8];
#pragma unroll
    for (int k = 1; k < 8; ++k) gs += red[8 + k];
    return gm + __logf(gs);
}

// Column-band LSE: column c sees rows j in [c-64, c-1] (clipped), entry
// st[(j-c+64)*256 + j] + u[j], plus the dustbin row term u[256].
__device__ __forceinline__ float colBandLse(const float* __restrict__ st,
                                            const float* __restrict__ us, int c) {
    int j0 = c - 64; if (j0 < 0) j0 = 0;
    int j1 = c - 1;  if (j1 > 255) j1 = 255;
    float ud = us[256];
    float m = ud;
    for (int j = j0; j <= j1; ++j)
        m = fmaxf(m, st[(j - c + 64) * 256 + j] + us[j]);
    float s = __expf(ud - m);
    for (int j = j0; j <= j1; ++j)
        s += __expf(st[(j - c + 64) * 256 + j] + us[j] - m);
    return m + __logf(s);
}

// Row-band LSE: row j sees d=0..63 at column c=j-d+64, plus dustbin col v[320].
__device__ __forceinline__ float rowBandLse(const float* __restrict__ st,
                                            const float* __restrict__ vs, int j) {
    float vd = vs[320];
    float m = vd;
#pragma unroll 4
    for (int d = 0; d < OT_D; ++d)
        m = fmaxf(m, st[d * 256 + j] + vs[j - d + 64]);
    float s = __expf(vd - m);
#pragma unroll 4
    for (int d = 0; d < OT_D; ++d)
        s += __expf(st[d * 256 + j] + vs[j - d + 64] - m);
    return m + __logf(s);
}

__global__ __launch_bounds__(256) void ot_disparity_sinkhorn(
    const float* __restrict__ scores,   // (B, D, H, W) f32
    const float* __restrict__ log_row,  // (W+1)
    const float* __restrict__ log_col,  // (C+1)
    float* __restrict__ out)            // (B, 1, H, W)
{
    extern __shared__ float smem[];
    float* st  = smem;                  // [64*256] band, transposed: st[d*256+j]
    float* us  = st + OT_D * OT_W;      // [257] (+pad)
    float* vs  = us + 260;              // [321] (+pad)
    float* lrs = vs + 324;              // [257] log_row
    float* lcs = lrs + 260;             // [321] log_col
    float* red = lcs + 324;             // [16] reduction scratch
    float* dns = red + 16;              // [256] denominators
    float* nms = dns + 256;             // [256] numerators

    const int tid  = threadIdx.x;
    const int wid  = tid >> 5;
    const int lane = tid & 31;
    const int bh   = blockIdx.x;        // b*64 + h
    const int b    = bh >> 6;
    const int h    = bh & 63;

    // --- marginals -> LDS ---
    for (int i = tid; i < OT_WP; i += 256) lrs[i] = log_row[i];
    for (int i = tid; i < OT_CP; i += 256) lcs[i] = log_col[i];

    // --- band scores -> LDS (transposed) via CDNA5 async global->LDS DMA ---
    // scores index = b*D*H*W + d*H*W + h*W + j ; LDS flat f = d*256 + j.
    {
        const float* src = scores + ((size_t)b * OT_D * OT_H + (size_t)h) * OT_W;
        unsigned lbase = (unsigned)(uintptr_t)(void*)st;  // low 32 bits = LDS offset
#pragma unroll
        for (int it = 0; it < 16; ++it) {
            int f = tid * 4 + it * 1024;       // 4 consecutive floats per lane
            int d = f >> 8;
            int j = f & 255;
            unsigned goff = (unsigned)((d * (OT_H * OT_W) + j) * 4);
            unsigned loff = lbase + (unsigned)(f * 4);
            asm volatile("global_load_async_to_lds_b128 %0, %1, %2 offset:0"
                         :: "v"(loff), "v"(goff),
                            "s"((unsigned long long)(uintptr_t)src)
                         : "memory");
        }
        asm volatile("s_wait_asynccnt 0x0" ::: "memory");
    }
    __syncthreads();

    // --- initial u0 = -LSE over columns (band + dustbin-col value 0) ---
    {
        const int j = tid;
        float m = 0.0f;                         // dustbin column entry is 0
        for (int d = 0; d < OT_D; ++d) m = fmaxf(m, st[d * 256 + j]);
        float s = __expf(-m);
        for (int d = 0; d < OT_D; ++d) s += __expf(st[d * 256 + j] - m);
        us[j] = -(m + __logf(s));
        if (tid == 0) us[256] = -__logf((float)OT_CP);  // dustbin row: all-zero row
    }
    __syncthreads();

    // --- Sinkhorn: 10 x (v-update, u-update) + final v-update ---
    for (int it = 0; it <= OT_ITERS; ++it) {
        // v-update: v[c] = log_col[c] - LSE_rows; v[320] uses LSE over all u.
        float lse_u = wgLse2(us[tid], us[256], tid == 0, red, wid, lane);
        float vn0 = lcs[tid] - colBandLse(st, us, tid);
        float vn1 = 0.0f;
        if (tid < 64) vn1 = lcs[256 + tid] - colBandLse(st, us, 256 + tid);
        vs[tid] = vn0;
        if (tid < 64) vs[256 + tid] = vn1;
        if (tid == 64) vs[320] = lcs[320] - lse_u;
        __syncthreads();
        if (it == OT_ITERS) break;              // reference ends on a v-update

        // u-update: u[j] = log_row[j] - LSE_cols; u[256] uses LSE over all v.
        float lse_v = wgLse2(vs[tid], (tid < 65) ? vs[256 + tid] : 0.0f,
                             tid < 65, red, wid, lane);
        float un = lrs[tid] - rowBandLse(st, vs, tid);
        us[tid] = un;
        if (tid == 0) us[256] = lrs[256] - lse_v;
        __syncthreads();
    }

    // --- Output: disparity expectation via WMMA dual-moment reduction ---
    // disp_map on the band equals d exactly, so per row j:
    //   denom = sum_d p[j][d],  numer = sum_d d*p[j][d],
    //   p = exp(s + u[j] + v[j-d+64]).
    // Per 16-row tile: chain 16 x v_wmma_f32_16x16x4_f32 over K=d with
    // B column 0 = ones (denominator), column 1 = d-values (numerator).
    {
        const int half = lane >> 4;            // A: lanes<16 carry K={0,1}, >=16 K={2,3}
        const int mm   = lane & 15;            // A row index M
        for (int pass = 0; pass < 2; ++pass) {
            const int rowbase = (wid * 2 + pass) * 16;
            const int j = rowbase + mm;
            const float uj = us[j];
            v8f acc = {};
            for (int t = 0; t < 16; ++t) {
                const int d0 = 4 * t + 2 * half;
                float p0 = __expf(st[d0 * 256 + j]       + uj + vs[j - d0 + 64]);
                float p1 = __expf(st[(d0 + 1) * 256 + j] + uj + vs[j - d0 + 63]);
                v2f a = {p0, p1};                           // A[M=mm][K=d0, d0+1]
                float b0 = (mm == 0) ? 1.0f : (mm == 1) ? (float)d0       : 0.0f;
                float b1 = (mm == 0) ? 1.0f : (mm == 1) ? (float)(d0 + 1) : 0.0f;
                v2f bb = {b0, b1};                          // B[K][N=mm]
                acc = __builtin_amdgcn_wmma_f32_16x16x4_f32(
                        false, a, false, bb, (short)0, acc, false, false);
            }
            // D layout: VGPR r, lanes<16 -> (M=r, N=lane); lanes>=16 -> (M=r+8).
            union { v8f v; float f[8]; } ua; ua.v = acc;
            if (mm == 0) {                      // N=0 column: denominators
#pragma unroll
                for (int r = 0; r < 8; ++r) dns[rowbase + half * 8 + r] = ua.f[r];
            } else if (mm == 1) {               // N=1 column: numerators
#pragma unroll
                for (int r = 0; r < 8; ++r) nms[rowbase + half * 8 + r] = ua.f[r];
            }
        }
    }
    __syncthreads();

    out[(size_t)bh * OT_W + tid] = nms[tid] / fmaxf(dns[tid], 1e-8f);
}

extern "C" void kernel_launch(void* const* d_in, const int* in_sizes, int n_in,
                              void* d_out, int out_size, void* d_ws, size_t ws_size,
                              hipStream_t stream) {
    (void)in_sizes; (void)n_in; (void)out_size; (void)d_ws; (void)ws_size;
    const float* scores  = (const float*)d_in[0];   // (2,64,64,256) f32
    const float* log_row = (const float*)d_in[1];   // (257)
    const float* log_col = (const float*)d_in[2];   // (321)
    float* out = (float*)d_out;                     // (2,1,64,256) f32

    const size_t smem_floats =
        (size_t)OT_D * OT_W + 260 + 324 + 260 + 324 + 16 + 256 + 256;
    const size_t smem_bytes = smem_floats * sizeof(float);  // ~72 KB (<320 KB/WGP)

    ot_disparity_sinkhorn<<<dim3(OT_B * OT_H), dim3(256), smem_bytes, stream>>>(
        scores, log_row, log_col, out);
}